// NodeGNNModel_58695023067297
// MI455X (gfx1250) — compile-verified
//
#include <hip/hip_runtime.h>

typedef __attribute__((ext_vector_type(2))) float v2f;
typedef __attribute__((ext_vector_type(8))) float v8f;

#define NFEAT_IN 128
#define HID1 64
#define HID2 32

// ---------------------------------------------------------------- utilities
__global__ void fill_f32(float* __restrict__ p, float v, long n) {
  long t = (long)blockIdx.x * blockDim.x + threadIdx.x;
  if (t < n) p[t] = v;
}

__global__ void degree_kernel(const int* __restrict__ dst, float* __restrict__ deg,
                              int n_edges) {
  int t = blockIdx.x * blockDim.x + threadIdx.x;
  if (t < n_edges) atomicAdd(&deg[dst[t]], 1.0f);
}

__global__ void rsqrt_kernel(float* __restrict__ d, int n) {
  int t = blockIdx.x * blockDim.x + threadIdx.x;
  if (t < n) d[t] = rsqrtf(d[t]);
}

// ---------------------------------------------------------------- WMMA GEMM
// H[n_rows x NOUT] = X[n_rows x K] @ W[K x NOUT], fp32 via V_WMMA_F32_16X16X4_F32.
// One wave32 computes one 16x16 output tile. All guards are wave-uniform or
// clamped so EXEC stays all-ones through the WMMA chain (ISA 7.12 requirement).
template <int K, int NOUT>
__global__ __launch_bounds__(256) void gemm_wmma(const float* __restrict__ X,
                                                 const float* __restrict__ W,
                                                 float* __restrict__ H, int n_rows) {
  const int lane = threadIdx.x & 31;
  const int wave = blockIdx.x * (blockDim.x >> 5) + (threadIdx.x >> 5);
  const int tilesN = NOUT >> 4;
  const int m0 = (wave / tilesN) << 4;
  const int n0 = (wave % tilesN) << 4;
  if (m0 >= n_rows) return;  // wave-uniform

  const int half = lane >> 4;  // 0: K=0,1  1: K=2,3 (per 32-bit A/B VGPR layout)
  const int l = lane & 15;
  const int mr = (m0 + l < n_rows) ? (m0 + l) : (n_rows - 1);  // clamp, keep EXEC full

  const float* __restrict__ xrow = X + (size_t)mr * K + 2 * half;       // A: row mr
  const float* __restrict__ wcol = W + (size_t)(2 * half) * NOUT + n0 + l;  // B: col n0+l

  v8f acc = {};
#pragma unroll
  for (int k = 0; k < K; k += 4) {
    v2f a;
    a.x = xrow[k];       // X[mr][k+2h]   (contiguous pair -> b64 load)
    a.y = xrow[k + 1];   // X[mr][k+2h+1]
    v2f b;
    b.x = wcol[(size_t)k * NOUT];         // W[k+2h][n0+l]
    b.y = wcol[(size_t)k * NOUT + NOUT];  // W[k+2h+1][n0+l]
    acc = __builtin_amdgcn_wmma_f32_16x16x4_f32(false, a, false, b, (short)0, acc,
                                                false, false);
  }

  // C/D layout: VGPR r -> row m0 + r + 8*half, col n0 + l
#pragma unroll
  for (int r = 0; r < 8; ++r) {
    int row = m0 + 8 * half + r;
    if (row < n_rows) H[(size_t)row * NOUT + n0 + l] = acc[r];
  }
}

// ---------------------------------------------------------------- edge scatter
// One thread per (edge, float4 chunk): gather h[src] (L2-resident), scale by
// norm = dinv[src]*dinv[dst], scatter-add into agg[dst] with f32 atomics.
template <int LOG2_NF4>  // NF = 4 << LOG2_NF4 features
__global__ void scatter_kernel(const int* __restrict__ src, const int* __restrict__ dst,
                               const float* __restrict__ dinv,
                               const float* __restrict__ h, float* __restrict__ agg,
                               int n_edges) {
  long tid = (long)blockIdx.x * blockDim.x + threadIdx.x;
  int e = (int)(tid >> LOG2_NF4);
  int c = (int)(tid & ((1 << LOG2_NF4) - 1));
  if (e >= n_edges) return;
  const int NF = 4 << LOG2_NF4;
  int s = src[e];
  int d = dst[e];
  float norm = dinv[s] * dinv[d];
  const float4* hs = (const float4*)(h + (size_t)s * NF);
  float4 v = hs[c];
  float* ad = agg + (size_t)d * NF + 4 * c;
  atomicAdd(ad + 0, v.x * norm);
  atomicAdd(ad + 1, v.y * norm);
  atomicAdd(ad + 2, v.z * norm);
  atomicAdd(ad + 3, v.w * norm);
}

// ---------------------------------------------------------------- epilogue 1
// agg[i,f] = relu(agg[i,f] + h[i,f]*dinv[i]^2 + bias[f])   (in place)
template <int LOG2_NF>
__global__ void finish_kernel(const float* __restrict__ h, const float* __restrict__ dinv,
                              const float* __restrict__ bias, float* __restrict__ agg,
                              int n_nodes) {
  long t = (long)blockIdx.x * blockDim.x + threadIdx.x;
  int i = (int)(t >> LOG2_NF);
  int f = (int)(t & ((1 << LOG2_NF) - 1));
  if (i >= n_nodes) return;
  float di = dinv[i];
  float v = agg[t] + h[t] * di * di + bias[f];
  agg[t] = v > 0.0f ? v : 0.0f;
}

// ---------------------------------------------------------------- final layer
// One wave32 per node; lane == feature (HID2 == 32). ReLU epilogue of layer 2
// fused with the [32 x 1] FC dot (shfl_xor wave reduction) + sigmoid.
__global__ void final_kernel(const float* __restrict__ h2, const float* __restrict__ agg2,
                             const float* __restrict__ dinv, const float* __restrict__ b2,
                             const float* __restrict__ Wfc, const float* __restrict__ bfc,
                             float* __restrict__ out, int n_nodes) {
  int lane = threadIdx.x & 31;
  int node = blockIdx.x * (blockDim.x >> 5) + (threadIdx.x >> 5);
  if (node >= n_nodes) return;
  float di = dinv[node];
  size_t idx = (size_t)node * HID2 + lane;
  float v = agg2[idx] + h2[idx] * di * di + b2[lane];
  v = v > 0.0f ? v : 0.0f;
  float p = v * Wfc[lane];
#pragma unroll
  for (int off = 16; off > 0; off >>= 1) p += __shfl_xor(p, off, 32);
  if (lane == 0) out[node] = 1.0f / (1.0f + __expf(-(p + bfc[0])));
}

// ---------------------------------------------------------------- launcher
extern "C" void kernel_launch(void* const* d_in, const int* in_sizes, int n_in,
                              void* d_out, int out_size, void* d_ws, size_t ws_size,
                              hipStream_t stream) {
  const float* x    = (const float*)d_in[0];
  const int*   eidx = (const int*)d_in[1];
  const float* W1   = (const float*)d_in[2];
  const float* b1   = (const float*)d_in[3];
  const float* W2   = (const float*)d_in[4];
  const float* b2   = (const float*)d_in[5];
  const float* Wfc  = (const float*)d_in[6];
  const float* bfc  = (const float*)d_in[7];
  float* out = (float*)d_out;

  const int N = in_sizes[0] / NFEAT_IN;
  const int E = in_sizes[1] / 2;
  const int* src = eidx;
  const int* dst = eidx + E;

  // workspace layout (fp32): dinv[N] | h1[N*64] | agg1[N*64] | h2[N*32] | agg2[N*32]
  float* dinv = (float*)d_ws;
  float* h1   = dinv + (((size_t)N + 63) & ~(size_t)63);
  float* agg1 = h1 + (size_t)N * HID1;
  float* h2   = agg1 + (size_t)N * HID1;
  float* agg2 = h2 + (size_t)N * HID2;

  const int B = 256;

  // degrees (self-loop => init to 1.0), then dinv = rsqrt(deg)
  fill_f32<<<(N + B - 1) / B, B, 0, stream>>>(dinv, 1.0f, N);
  fill_f32<<<(int)(((long)N * HID1 + B - 1) / B), B, 0, stream>>>(agg1, 0.0f, (long)N * HID1);
  fill_f32<<<(int)(((long)N * HID2 + B - 1) / B), B, 0, stream>>>(agg2, 0.0f, (long)N * HID2);
  degree_kernel<<<(E + B - 1) / B, B, 0, stream>>>(dst, dinv, E);
  rsqrt_kernel<<<(N + B - 1) / B, B, 0, stream>>>(dinv, N);

  // ---- layer 1: h1 = x @ W1 ; scatter ; relu epilogue (into agg1) ----
  {
    int tiles = ((N + 15) / 16) * (HID1 / 16);
    gemm_wmma<NFEAT_IN, HID1><<<(tiles + 7) / 8, 256, 0, stream>>>(x, W1, h1, N);
  }
  {
    long total = (long)E * (HID1 / 4);
    scatter_kernel<4><<<(int)((total + B - 1) / B), B, 0, stream>>>(src, dst, dinv, h1,
                                                                    agg1, E);
  }
  {
    long total = (long)N * HID1;
    finish_kernel<6><<<(int)((total + B - 1) / B), B, 0, stream>>>(h1, dinv, b1, agg1, N);
  }

  // ---- layer 2: h2 = agg1 @ W2 ; scatter ; fused epilogue + FC + sigmoid ----
  {
    int tiles = ((N + 15) / 16) * (HID2 / 16);
    gemm_wmma<HID1, HID2><<<(tiles + 7) / 8, 256, 0, stream>>>(agg1, W2, h2, N);
  }
  {
    long total = (long)E * (HID2 / 4);
    scatter_kernel<3><<<(int)((total + B - 1) / B), B, 0, stream>>>(src, dst, dinv, h2,
                                                                    agg2, E);
  }
  final_kernel<<<(N + 7) / 8, 256, 0, stream>>>(h2, agg2, dinv, b2, Wfc, bfc, out, N);
}